// GaussianRenderer_68573447847947
// MI455X (gfx1250) — compile-verified
//
#include <hip/hip_runtime.h>
#include <hip/hip_bf16.h>

typedef __attribute__((ext_vector_type(2))) float v2f;
typedef __attribute__((ext_vector_type(8))) float v8f;

#define N_GAUSS 256
#define IMG 256
#define REC 12   // floats per gaussian record

// ---------------------------------------------------------------------------
// Preprocess: project (WMMA f32 16x16x4), build per-gaussian records,
// rank-sort by camera z (descending, stable), write sorted AoS to ws.
// Record: [xc, yc, 1/xs, 1/ys, xmin, xmax, ymin, ymax, r, g, b, op]
// One block per batch, 256 threads (8 waves, each wave -> 2 tiles of 16).
// ---------------------------------------------------------------------------
__global__ __launch_bounds__(256)
void GaussianRenderer_prep(const float* __restrict__ pos,     // (N,3)
                           const float* __restrict__ scales,  // (N,3)
                           const float* __restrict__ colors,  // (N,3)
                           const float* __restrict__ opac,    // (N,1)
                           const float* __restrict__ view,    // (B,4,4)
                           const float* __restrict__ proj,    // (B,4,4)
                           float* __restrict__ ws)            // (B,256,12)
{
    const int b    = blockIdx.x;
    const int t    = threadIdx.x;
    const int lane = t & 31;
    const int wv   = t >> 5;

    __shared__ float sKey[N_GAUSS];
    __shared__ float sRec[N_GAUSS * REC];

    const float* V  = view + b * 16;
    const float* Pm = proj + b * 16;

    // A operand: combined matrix rows. Row r(0..3) = (proj@view) row r,
    // row 4 = view row 2 (camera z), rows 5..15 = 0.
    // ISA 16x4 f32 A layout: lane<16 -> M=lane, K=0(VGPR0),1(VGPR1);
    //                        lane>=16 -> M=lane-16, K=2(VGPR0),3(VGPR1).
    const int r  = lane & 15;
    const int hi = lane >> 4;        // 0 -> K={0,1}, 1 -> K={2,3}
    const int k0 = hi * 2, k1 = hi * 2 + 1;
    float a0 = 0.f, a1 = 0.f;
    if (r < 4) {
        float p0 = 0.f, p1 = 0.f;
        for (int j = 0; j < 4; ++j) {
            p0 += Pm[r * 4 + j] * V[j * 4 + k0];
            p1 += Pm[r * 4 + j] * V[j * 4 + k1];
        }
        a0 = p0; a1 = p1;
    } else if (r == 4) {
        a0 = V[2 * 4 + k0]; a1 = V[2 * 4 + k1];
    }
    v2f A = {a0, a1};

    // B operand: hom positions of 16 gaussians. B[k][n] = hom_k(g_base+n).
    // Assumed layout mirrors A striping: lane<16 -> K=0,1 ; lane>=16 -> K=2,3.
    const int n  = lane & 15;
    const int g0 = wv * 32 + n;          // tile 2*wv
    const int g1 = wv * 32 + 16 + n;     // tile 2*wv+1
    float bx0 = (lane < 16) ? pos[g0 * 3 + 0] : pos[g0 * 3 + 2];
    float by0 = (lane < 16) ? pos[g0 * 3 + 1] : 1.0f;
    float bx1 = (lane < 16) ? pos[g1 * 3 + 0] : pos[g1 * 3 + 2];
    float by1 = (lane < 16) ? pos[g1 * 3 + 1] : 1.0f;
    v2f B0 = {bx0, by0};
    v2f B1 = {bx1, by1};

    v8f c = {};
    // D[m][n]: m=output component, n=gaussian-in-tile.
    // Lane l<16 holds components 0..7 (VGPR 0..7) of gaussian base+l.
    v8f d0 = __builtin_amdgcn_wmma_f32_16x16x4_f32(false, A, false, B0,
                                                   (short)0, c, false, false);
    v8f d1 = __builtin_amdgcn_wmma_f32_16x16x4_f32(false, A, false, B1,
                                                   (short)0, c, false, false);

    if (lane < 16) {
        for (int s = 0; s < 2; ++s) {
            v8f d  = s ? d1 : d0;
            int g  = s ? g1 : g0;
            float pw   = fmaxf(d[3], 1e-8f);
            float xc   = (d[0] / pw + 1.0f) * (0.5f * (float)IMG);
            float yc   = (d[1] / pw + 1.0f) * (0.5f * (float)IMG);
            float xs   = scales[g * 3 + 0] * (float)IMG;
            float ys   = scales[g * 3 + 1] * (float)IMG;
            float xmin = floorf(fmaxf(xc - 3.0f * xs, 0.0f));
            float xmax = fminf(truncf(xc + 3.0f * xs), (float)IMG);
            float ymin = floorf(fmaxf(yc - 3.0f * ys, 0.0f));
            float ymax = fminf(truncf(yc + 3.0f * ys), (float)IMG);
            sKey[g] = d[4];   // camera z
            float* rec = &sRec[g * REC];
            rec[0] = xc;   rec[1] = yc;
            rec[2] = 1.0f / xs;   rec[3] = 1.0f / ys;
            rec[4] = xmin; rec[5] = xmax; rec[6] = ymin; rec[7] = ymax;
            rec[8]  = colors[g * 3 + 0];
            rec[9]  = colors[g * 3 + 1];
            rec[10] = colors[g * 3 + 2];
            rec[11] = opac[g];
        }
    }
    __syncthreads();

    // Stable rank sort: position in descending-camz order (argsort(-z), stable).
    float key  = sKey[t];
    int   rank = 0;
    for (int j = 0; j < N_GAUSS; ++j) {
        float kj = sKey[j];
        rank += (kj > key) || ((kj == key) && (j < t)) ? 1 : 0;
    }
    float* dst = ws + ((size_t)b * N_GAUSS + rank) * REC;
    const float* rec = &sRec[t * REC];
    for (int k = 0; k < REC; ++k) dst[k] = rec[k];
}

// ---------------------------------------------------------------------------
// Render: one thread per pixel, 16x16 pixel tile per block.
// Sorted records async-DMA'd into LDS (global_load_async_to_lds_b128,
// ASYNCcnt), then sequential front-to-back composite over 256 gaussians.
// ---------------------------------------------------------------------------
__global__ __launch_bounds__(256)
void GaussianRenderer_render(const float* __restrict__ ws,   // (B,256,12)
                             float* __restrict__ out)        // (B,S,S,3)
{
    const int b    = blockIdx.y;
    const int tile = blockIdx.x;            // 0..255 : 16x16 tiles
    const int t    = threadIdx.x;           // 0..255
    const int px   = (tile & 15) * 16 + (t & 15);
    const int py   = (tile >> 4) * 16 + (t >> 4);

    __shared__ float4 sg[N_GAUSS * 3];      // 12 KB, 16B-aligned records

    // LDS byte offset = low 32 bits of the generic (LDS-aperture) address.
    unsigned lds_base = (unsigned)(unsigned long long)(void*)sg;
    const float* src  = ws + (size_t)b * N_GAUSS * REC;

    // 3072 dwords = 768 x b128 ; 256 lanes x 3 transfers.
    for (int i = 0; i < 3; ++i) {
        unsigned off      = ((unsigned)t + (unsigned)i * 256u) * 16u;
        unsigned lds_addr = lds_base + off;
        asm volatile("global_load_async_to_lds_b128 %0, %1, %2"
                     :
                     : "v"(lds_addr), "v"(off), "s"(src)
                     : "memory");
    }
    asm volatile("s_wait_asynccnt 0" ::: "memory");
    __syncthreads();

    const float fx = (float)px;
    const float fy = (float)py;
    float imR = 0.f, imG = 0.f, imB = 0.f, Aacc = 0.f;

    for (int g = 0; g < N_GAUSS; ++g) {
        float4 r0 = sg[g * 3 + 0];   // xc, yc, 1/xs, 1/ys
        float4 r1 = sg[g * 3 + 1];   // xmin, xmax, ymin, ymax
        float4 r2 = sg[g * 3 + 2];   // r, g, b, op
        float dx = (fx - r0.x) * r0.z;
        float dy = (fy - r0.y) * r0.w;
        bool inb = (fx >= r1.x) & (fx < r1.y) & (fy >= r1.z) & (fy < r1.w);
        float e  = __expf(-0.5f * (dx * dx + dy * dy)) * r2.w;
        float gv = inb ? e : 0.0f;
        float alpha = gv * (1.0f - Aacc);
        float w2    = alpha * gv;
        imR += w2 * r2.x;
        imG += w2 * r2.y;
        imB += w2 * r2.z;
        Aacc += alpha;
    }

    float* o = out + (((size_t)b * IMG + py) * IMG + px) * 3;
    o[0] = imR; o[1] = imG; o[2] = imB;
}

// ---------------------------------------------------------------------------
extern "C" void kernel_launch(void* const* d_in, const int* in_sizes, int n_in,
                              void* d_out, int out_size, void* d_ws, size_t ws_size,
                              hipStream_t stream) {
    const float* positions = (const float*)d_in[0];
    const float* scales    = (const float*)d_in[1];
    // d_in[2] = rotations (unused by reference math)
    const float* colors    = (const float*)d_in[3];
    const float* opac      = (const float*)d_in[4];
    const float* view      = (const float*)d_in[5];
    const float* proj      = (const float*)d_in[6];
    float* ws  = (float*)d_ws;            // 2 * 256 * 12 floats = 24 KB
    float* out = (float*)d_out;

    GaussianRenderer_prep<<<dim3(2), 256, 0, stream>>>(
        positions, scales, colors, opac, view, proj, ws);
    GaussianRenderer_render<<<dim3(256, 2), 256, 0, stream>>>(ws, out);
}